// Summarizer_74741020885454
// MI455X (gfx1250) — compile-verified
//
#include <hip/hip_runtime.h>
#include <hip/hip_bf16.h>

// Shapes from the reference
#define B_  32
#define S_  4096
#define H_  768
#define N_  128

#define ROWS 16            // outputs per workgroup (M tile)
#define THREADS 128        // 4 waves of 32
#define WAVES 4
#define KPW (H_ / WAVES)   // 192 K-elements per wave (48 WMMA steps of K=4)

typedef float v2f __attribute__((ext_vector_type(2)));
typedef float v8f __attribute__((ext_vector_type(8)));

#if __has_builtin(__builtin_amdgcn_tensor_load_to_lds)
#define USE_TDM 1
#else
#define USE_TDM 0
#endif

#if USE_TDM
// TDM pad engine: 1 dword pad per 256 dwords stored -> row stride 771 dwords.
// 771 % 64 == 3 -> 16 A-fragment lanes land on 16 distinct LDS banks.
#define ROW_STRIDE 771
#define AOFF(m, k) ((m) * ROW_STRIDE + (k) + ((k) >> 8))
typedef unsigned int u32x4 __attribute__((ext_vector_type(4)));
typedef int          i32x4 __attribute__((ext_vector_type(4)));
typedef int          i32x8 __attribute__((ext_vector_type(8)));
#else
#define ROW_STRIDE 772     // manual pad: 772 % 64 == 4, also conflict-free
#define AOFF(m, k) ((m) * ROW_STRIDE + (k))
#endif

__global__ __launch_bounds__(THREADS) void summarizer_gather_wmma(
    const float* __restrict__ enc,   // [B, S, H] fp32
    const int*   __restrict__ idx,   // [B, N] int32
    const float* __restrict__ W,     // [H] fp32
    const float* __restrict__ bias,  // [1] fp32
    float*       __restrict__ out)   // [B, N] fp32
{
    __shared__ float ldsA[ROWS * ROW_STRIDE]; // 16 gathered rows (padded)
    __shared__ float ldsW[H_];                // classifier weights
    __shared__ float psum[WAVES][32][8];      // per-wave WMMA accumulators

    const int tid  = threadIdx.x;
    const int lane = tid & 31;
    const int wave = tid >> 5;

    const int g  = blockIdx.x;        // 0 .. B*N/16 - 1  (256)
    const int b  = g >> 3;            // 8 tiles per batch (N=128 / 16)
    const int n0 = (g & 7) * ROWS;

    // Stage W into LDS (broadcast-read later by both half-waves).
    for (int i = tid; i < H_; i += THREADS) ldsW[i] = W[i];

#if USE_TDM
    // ---- Tensor Data Mover, gather mode: DMA 16 indexed rows into LDS ----
    if (wave == 0) {
        // Uniform (SGPR) loads of the 16 sorted row indices for this tile.
        const int* ip = idx + b * N_ + n0;
        unsigned long long gaddr =
            (unsigned long long)(uintptr_t)(enc + (size_t)b * S_ * H_);
        unsigned int lds_base = (unsigned int)(uintptr_t)&ldsA[0]; // flat low 32 = LDS offset

        u32x4 g0;
        g0[0] = 0x80000001u;            // count=1 | gather_mode=1 | 16-bit indices
        g0[1] = lds_base;               // [63:32] lds_addr
        g0[2] = (unsigned int)(gaddr & 0xFFFFFFFFu);            // global_addr lo
        g0[3] = (unsigned int)((gaddr >> 32) & 0x01FFFFFFu)     // global_addr[56:32]
                | 0x80000000u;                                  // type=2 ("image")

        i32x8 g1;
        g1[0] = (2 << 16)      // data_size = 4B
              | (1 << 20)      // pad_enable
              | (7 << 22)      // pad_interval: 256 dwords
              | (0 << 25);     // pad_amount: 1 dword
        g1[1] = (int)(768u  << 16);   // tensor_dim0 lo16 @ [63:48]
        g1[2] = (int)(4096u << 16);   // tensor_dim0 hi16=0, tensor_dim1 lo16 @ [95:80]
        g1[3] = (int)(768u  << 16);   // tensor_dim1 hi16=0, tile_dim0 @ [127:112]
        g1[4] = ROWS;                 // tile_dim1 = #valid gather indices @ [143:128]
        g1[5] = 768;                  // tensor_dim0_stride lo32
        g1[6] = 0;                    // dim0_stride hi16, dim1_stride lo16 (ignored)
        g1[7] = 0;

        i32x4 g2, g3;                 // 16-bit row indices, 2 per dword
        g2[0] = (ip[0]  & 0xFFFF) | (ip[1]  << 16);
        g2[1] = (ip[2]  & 0xFFFF) | (ip[3]  << 16);
        g2[2] = (ip[4]  & 0xFFFF) | (ip[5]  << 16);
        g2[3] = (ip[6]  & 0xFFFF) | (ip[7]  << 16);
        g3[0] = (ip[8]  & 0xFFFF) | (ip[9]  << 16);
        g3[1] = (ip[10] & 0xFFFF) | (ip[11] << 16);
        g3[2] = (ip[12] & 0xFFFF) | (ip[13] << 16);
        g3[3] = (ip[14] & 0xFFFF) | (ip[15] << 16);

        i32x8 gx = {0, 0, 0, 0, 0, 0, 0, 0}; // extra arg of the 6-arg clang-23 form

        __builtin_amdgcn_tensor_load_to_lds(g0, g1, g2, g3, gx, 0);
        __builtin_amdgcn_s_wait_tensorcnt(0);   // LDS data landed
    }
    __syncthreads();
#else
    // ---- Fallback: cooperative coalesced float4 gather ----
    if (tid < ROWS) {
        int r = idx[b * N_ + n0 + tid];
        __builtin_prefetch(&enc[((size_t)b * S_ + (size_t)r) * (size_t)H_], 0, 0);
    }
    for (int j = tid; j < ROWS * (H_ / 4); j += THREADS) {
        int row = j / (H_ / 4);
        int c4  = j % (H_ / 4);
        int r   = idx[b * N_ + n0 + row];
        const float4* src =
            (const float4*)(enc + ((size_t)b * S_ + (size_t)r) * (size_t)H_);
        float4 v = src[c4];
        float* dst = &ldsA[row * ROW_STRIDE + c4 * 4];
        dst[0] = v.x; dst[1] = v.y; dst[2] = v.z; dst[3] = v.w;
    }
    __syncthreads();
#endif

    // ---- Split-K fp32 WMMA: wave w covers K in [w*192, w*192+192) ----
    // A fragment (16x4 f32, ISA layout): lane = m + 16*(k>=2), 2 floats/lane.
    // B fragment (4x16): W[k] replicated across all 16 columns -> every
    // column of D equals the desired dot product.
    const int m    = lane & 15;
    const int koff = (lane < 16) ? 0 : 2;

    v8f c = {};
    const int kbase0 = wave * KPW;
    #pragma unroll 4
    for (int kk = 0; kk < KPW; kk += 4) {
        int k = kbase0 + kk + koff;
        v2f a;
        a.x = ldsA[AOFF(m, k)];
        a.y = ldsA[AOFF(m, k) + 1];           // same pad region: k even, pads at 256-dword marks
        v2f bfrag = *(const v2f*)&ldsW[k];    // k even -> 8B aligned
        // 8 args: (neg_a, A, neg_b, B, c_mod, C, reuse_a, reuse_b)
        c = __builtin_amdgcn_wmma_f32_16x16x4_f32(
                false, a, false, bfrag, (short)0, c, false, false);
    }

    // Cross-wave split-K reduction through LDS.
    #pragma unroll
    for (int r = 0; r < 8; ++r) psum[wave][lane][r] = c[r];
    __syncthreads();

    if (wave == 0) {
        float acc[8];
        #pragma unroll
        for (int r = 0; r < 8; ++r)
            acc[r] = psum[0][lane][r] + psum[1][lane][r] +
                     psum[2][lane][r] + psum[3][lane][r];
        float bv = bias[0];
        // D layout: VGPR r, lanes 0-15 -> M=r (any N); lanes 16-31 -> M=8+r.
        if (lane == 0) {
            #pragma unroll
            for (int r = 0; r < 8; ++r) out[b * N_ + n0 + r] = acc[r] + bv;
        } else if (lane == 16) {
            #pragma unroll
            for (int r = 0; r < 8; ++r) out[b * N_ + n0 + 8 + r] = acc[r] + bv;
        }
    }
}

extern "C" void kernel_launch(void* const* d_in, const int* in_sizes, int n_in,
                              void* d_out, int out_size, void* d_ws, size_t ws_size,
                              hipStream_t stream) {
    const float* enc  = (const float*)d_in[0]; // [B,S,H]
    const int*   idx  = (const int*)  d_in[1]; // [B,N]
    const float* W    = (const float*)d_in[2]; // [H,1]
    const float* bias = (const float*)d_in[3]; // [1]
    float* out = (float*)d_out;                // [B,N]

    dim3 grid((B_ * N_) / ROWS);               // 256 workgroups
    dim3 block(THREADS);                       // 128 threads = 4 wave32
    summarizer_gather_wmma<<<grid, block, 0, stream>>>(enc, idx, W, bias, out);
}